// GridSamplePScan_19602230739403
// MI455X (gfx1250) — compile-verified
//
#include <hip/hip_runtime.h>
#include <stdint.h>

// Problem constants (from the reference): B=4, L=32, C=32, H=W=128.
#define BB 4
#define LL 32
#define CC 32
#define HH 128
#define WW 128
#define HWPIX (HH * WW)          // 16384 pixels per plane

// ---------------------------------------------------------------------------
// Compose kernel: out[l] = cur[l] + bilinear_sample(prev[l-step], grid(cur_flow[l]))
// Grid: x = pixel tiles, y = slice index (lq - step), z = batch. No integer
// division anywhere. Coords/weights computed once per pixel, reused for
// 2 flow + 32 image channels. Clamped gathers with zeroed weights reproduce
// the reference's zero-padded, validity-masked gather exactly.
// Read-once streams (current flow/image) use non-temporal loads so L0/L2
// capacity is reserved for the spatially-reused gather taps.
// ---------------------------------------------------------------------------
__global__ __launch_bounds__(256) void compose_step(
    const float* __restrict__ src_f, const float* __restrict__ src_i,
    float* __restrict__ dst_f, float* __restrict__ dst_i, int step)
{
    const int b  = blockIdx.z;
    const int lq = step + blockIdx.y;       // "current" slice
    const int lp = lq - step;               // "previous" slice

    const int idx = blockIdx.x * blockDim.x + threadIdx.x;   // pixel 0..HW-1
    const int h = idx >> 7;                 // / 128
    const int w = idx & 127;                // % 128

    const size_t qf_off = ((size_t)(b * LL + lq) * 2) * HWPIX;
    const size_t pf_off = ((size_t)(b * LL + lp) * 2) * HWPIX;
    const size_t qi_off = ((size_t)(b * LL + lq) * CC) * HWPIX;
    const size_t pi_off = ((size_t)(b * LL + lp) * CC) * HWPIX;

    // --- grid generation (matches _gen_grid) ---
    const float fqx = __builtin_nontemporal_load(&src_f[qf_off + idx]);
    const float fqy = __builtin_nontemporal_load(&src_f[qf_off + HWPIX + idx]);
    const float stepx = 2.0f / WW, stepy = 2.0f / HH;
    const float gx = w * stepx + (-1.0f + 0.5f * stepx);
    const float gy = h * stepy + (-1.0f + 0.5f * stepy);
    float fx = gx + fqx;
    float fy = gy + fqy;
    // wrap x: remainder(fx+1, 2) - 1   (jnp.remainder -> result in [0,2))
    float t = fx + 1.0f;
    t = t - floorf(t * 0.5f) * 2.0f;
    fx = t - 1.0f;

    // --- normalized -> pixel coords (align_corners=False) ---
    const float x = (fx + 1.0f) * (WW * 0.5f) - 0.5f;
    const float y = (fy + 1.0f) * (HH * 0.5f) - 0.5f;
    const float x0f = floorf(x), y0f = floorf(y);
    const float tx = x - x0f,    ty = y - y0f;
    const int x0 = (int)x0f, y0 = (int)y0f;
    const int x1 = x0 + 1,   y1 = y0 + 1;

    float w00 = (1.0f - tx) * (1.0f - ty);
    float w10 = tx * (1.0f - ty);
    float w01 = (1.0f - tx) * ty;
    float w11 = tx * ty;

    const bool vx0 = (x0 >= 0) & (x0 < WW);
    const bool vx1 = (x1 >= 0) & (x1 < WW);
    const bool vy0 = (y0 >= 0) & (y0 < HH);
    const bool vy1 = (y1 >= 0) & (y1 < HH);
    w00 = (vx0 && vy0) ? w00 : 0.0f;
    w10 = (vx1 && vy0) ? w10 : 0.0f;
    w01 = (vx0 && vy1) ? w01 : 0.0f;
    w11 = (vx1 && vy1) ? w11 : 0.0f;

    const int cx0 = min(max(x0, 0), WW - 1);
    const int cx1 = min(max(x1, 0), WW - 1);
    const int cy0 = min(max(y0, 0), HH - 1);
    const int cy1 = min(max(y1, 0), HH - 1);
    const int o00 = cy0 * WW + cx0;
    const int o10 = cy0 * WW + cx1;
    const int o01 = cy1 * WW + cx0;
    const int o11 = cy1 * WW + cx1;

    // --- flow channels (2) ---
    #pragma unroll
    for (int c = 0; c < 2; ++c) {
        const float* pf = src_f + pf_off + (size_t)c * HWPIX;
        const float qv = __builtin_nontemporal_load(&src_f[qf_off + (size_t)c * HWPIX + idx]);
        const float s = w00 * pf[o00] + w10 * pf[o10] + w01 * pf[o01] + w11 * pf[o11];
        dst_f[qf_off + (size_t)c * HWPIX + idx] = qv + s;
    }
    // --- image channels (32) ---
    #pragma unroll 8
    for (int c = 0; c < CC; ++c) {
        const float* pi = src_i + pi_off + (size_t)c * HWPIX;
        const float qv = __builtin_nontemporal_load(&src_i[qi_off + (size_t)c * HWPIX + idx]);
        const float s = w00 * pi[o00] + w10 * pi[o10] + w01 * pi[o01] + w11 * pi[o11];
        dst_i[qi_off + (size_t)c * HWPIX + idx] = qv + s;
    }
}

// ---------------------------------------------------------------------------
// TDM bulk copy: pass-through slices l < step are pure DMA. One wave per
// 64 KB chunk builds D# descriptors in SGPRs and drives the Tensor Data
// Mover: global -> LDS -> global, ordered with s_wait_tensorcnt.
// TDM ignores EXEC, so the block is exactly one wave (32 threads) to issue
// each DMA exactly once.
// ---------------------------------------------------------------------------
typedef __attribute__((ext_vector_type(4))) unsigned int u32x4;
typedef __attribute__((ext_vector_type(8))) unsigned int u32x8;

#define CHUNK_ELEMS 16384   // 64 KB of f32 per DMA chunk (fits 16-bit tile_dim0)

__global__ __launch_bounds__(32) void tdm_copy_slices(
    const float* __restrict__ src, float* __restrict__ dst,
    int planes /* channels per slice */, int nslices /* = step */)
{
    __shared__ __align__(16) char lds_buf[CHUNK_ELEMS * 4];

    const int chunk = blockIdx.x;
    const int b     = blockIdx.z;
    const int l     = blockIdx.y;           // l < step

    const size_t sliceElems = (size_t)planes * HWPIX;
    const size_t off = (size_t)(b * LL + l) * sliceElems + (size_t)chunk * CHUNK_ELEMS;

    const unsigned long long gsrc = (unsigned long long)(uintptr_t)(src + off);
    const unsigned long long gdst = (unsigned long long)(uintptr_t)(dst + off);
    // Low 32 bits of a generic LDS pointer == workgroup-relative LDS byte offset.
    const unsigned int lds = (unsigned int)(uintptr_t)(void*)lds_buf;
    const unsigned int elems = CHUNK_ELEMS;
    (void)nslices;

    // ---- D# group 0 (128b): count=1 | lds_addr | global_addr[56:0] | type=2 ----
    u32x4 g0l, g0s;
    g0l.x = 1u;                                   // count=1, user mode, no gather
    g0l.y = lds;                                  // LDS byte address
    g0l.z = (unsigned int)gsrc;                   // global_addr[31:0]
    g0l.w = ((unsigned int)(gsrc >> 32) & 0x01FFFFFFu) | 0x80000000u; // addr[56:32] | type=2
    g0s = g0l;
    g0s.z = (unsigned int)gdst;
    g0s.w = ((unsigned int)(gdst >> 32) & 0x01FFFFFFu) | 0x80000000u;

    // ---- D# group 1 (256b): data_size=4B, 1-D tensor/tile of `elems` ----
    u32x8 g1;
    g1.s0 = 0x00020000u;                          // workgroup_mask=0, data_size=2 (4B)
    g1.s1 = (elems & 0xFFFFu) << 16;              // tensor_dim0[15:0] (bits 79:48)
    g1.s2 = ((elems >> 16) & 0xFFFFu)             // tensor_dim0[31:16]
          | (1u << 16);                           // tensor_dim1 = 1
    g1.s3 = (elems & 0xFFFFu) << 16;              // tile_dim0 (bits 127:112)
    g1.s4 = 0u;                                   // tile_dim1=0 (unused), tile_dim2=0
    g1.s5 = elems;                                // tensor_dim0_stride[31:0]
    g1.s6 = 0u;
    g1.s7 = 0u;

    asm volatile(
        "tensor_load_to_lds %0, %1\n\t"
        "s_wait_tensorcnt 0x0\n\t"
        "tensor_store_from_lds %2, %3\n\t"
        "s_wait_tensorcnt 0x0"
        :
        : "s"(g0l), "s"(g1), "s"(g0s), "s"(g1)
        : "memory");
}

// ---------------------------------------------------------------------------
// Host-side orchestration: 5 doubling steps, ping-pong buffers.
// Workspace layout: flowsA (16MB) | flowsB (16MB) | imagesY (256MB).
// imagesX = d_out; steps 1,3,5 write X so the final result lands in d_out.
// ---------------------------------------------------------------------------
extern "C" void kernel_launch(void* const* d_in, const int* in_sizes, int n_in,
                              void* d_out, int out_size, void* d_ws, size_t ws_size,
                              hipStream_t stream)
{
    const float* in_f = (const float*)d_in[0];   // flows  [B,L,2,H,W]
    const float* in_i = (const float*)d_in[1];   // images [B,L,C,H,W]
    float* out = (float*)d_out;

    const size_t fElems = (size_t)BB * LL * 2 * HWPIX;    // 4,194,304
    char* ws = (char*)d_ws;
    float* fA = (float*)ws;
    float* fB = fA + fElems;
    float* iY = (float*)(ws + 2 * fElems * sizeof(float));
    float* iX = out;

    const float* cf = in_f;
    const float* ci = in_i;

    int s = 1;
    for (int k = 0; k < 5; ++k, s *= 2) {
        float* df = (k % 2 == 0) ? fA : fB;
        float* di = (k % 2 == 0) ? iX : iY;

        // Pass-through slices l < s : TDM DMA copies (flows then images).
        {
            dim3 gF((2 * HWPIX) / CHUNK_ELEMS, s, BB);      // 2 chunks per flow slice
            tdm_copy_slices<<<gF, 32, 0, stream>>>(cf, df, 2, s);
            dim3 gI((CC * HWPIX) / CHUNK_ELEMS, s, BB);     // 32 chunks per image slice
            tdm_copy_slices<<<gI, 32, 0, stream>>>(ci, di, CC, s);
        }
        // Composed slices l >= s.
        {
            dim3 gC(HWPIX / 256, LL - s, BB);
            compose_step<<<gC, 256, 0, stream>>>(cf, ci, df, di, s);
        }
        cf = df;
        ci = di;
    }
    // After k=4 (s=16), di == iX == d_out: final images are in d_out.
}